// MorphologyEx_75574244540569
// MI455X (gfx1250) — compile-verified
//
#include <hip/hip_runtime.h>
#include <cstdint>
#include <cstddef>

// Morphological opening (3x3 flat SE), fused erode+dilate, single HBM pass.
// (16,3,1024,1024) f32 -> 48 planes of 1024x1024.
// Roofline: ~402 MB moved @ 23.3 TB/s -> ~17 us floor; compute negligible,
// so v2 vectorizes all LDS stage traffic (b128) to stay bandwidth-bound.
//
// Tile: 128x16 outputs per 256-thread block (8 wave32 waves).
// LDS: two 20x136 f32 ping-pong buffers (halo 2; columns padded/aligned so
// the raw tile is staged with CDNA5 async global->LDS DMA, ASYNCcnt-tracked).

#define TX 128
#define TY 16
#define RW 136              // LDS row stride in floats (covers x0-4 .. x0+131)
#define RH 20               // covers y0-2 .. y0+17
#define IMG 1024

typedef __attribute__((address_space(3))) float lds_f;
typedef float v4f __attribute__((ext_vector_type(4)));

__device__ __forceinline__ v4f vmin3(v4f a, v4f b, v4f c) {
    v4f o;
    o.x = fminf(fminf(a.x, b.x), c.x);
    o.y = fminf(fminf(a.y, b.y), c.y);
    o.z = fminf(fminf(a.z, b.z), c.z);
    o.w = fminf(fminf(a.w, b.w), c.w);
    return o;
}
__device__ __forceinline__ v4f vmax3(v4f a, v4f b, v4f c) {
    v4f o;
    o.x = fmaxf(fmaxf(a.x, b.x), c.x);
    o.y = fmaxf(fmaxf(a.y, b.y), c.y);
    o.z = fmaxf(fmaxf(a.z, b.z), c.z);
    o.w = fmaxf(fmaxf(a.w, b.w), c.w);
    return o;
}
// horizontal 3-tap from shifted quad: out[k] = op(q[k-1], q[k], q[k+1])
__device__ __forceinline__ v4f hmin3(float l, v4f q, float r) {
    v4f o;
    o.x = fminf(fminf(l,   q.x), q.y);
    o.y = fminf(fminf(q.x, q.y), q.z);
    o.z = fminf(fminf(q.y, q.z), q.w);
    o.w = fminf(fminf(q.z, q.w), r);
    return o;
}
__device__ __forceinline__ v4f hmax3(float l, v4f q, float r) {
    v4f o;
    o.x = fmaxf(fmaxf(l,   q.x), q.y);
    o.y = fmaxf(fmaxf(q.x, q.y), q.z);
    o.z = fmaxf(fmaxf(q.y, q.z), q.w);
    o.w = fmaxf(fmaxf(q.z, q.w), r);
    return o;
}

__global__ __launch_bounds__(256)
void MorphOpen_f32_kernel(const float* __restrict__ in, float* __restrict__ out) {
    __shared__ __align__(16) float A[RH * RW];   // raw tile, later h-min (erosion)
    __shared__ __align__(16) float B[RH * RW];   // v-min, later v-max

    const int tid = threadIdx.x;
    const int x0  = blockIdx.x * TX;
    const int y0  = blockIdx.y * TY;
    const size_t plane = (size_t)blockIdx.z * (size_t)(IMG * IMG);
    const float* __restrict__ ip = in  + plane;
    float*       __restrict__ op = out + plane;
    const float PINF = __builtin_inff();

    // ---- Stage 0: stage raw tile (+inf outside image) into LDS -------------
    // 20 rows x 34 aligned float4 quads; in-bounds quads via async DMA.
    for (int i = tid; i < RH * 34; i += 256) {
        const int r  = i / 34;
        const int q  = i - r * 34;
        const int gy = y0 - 2 + r;
        const int gx = x0 - 4 + q * 4;            // multiple of 4 -> 16B aligned
        const int li = r * RW + q * 4;
        if ((unsigned)gy < (unsigned)IMG && (unsigned)gx <= (unsigned)(IMG - 4)) {
            const unsigned lo = (unsigned)(uintptr_t)((lds_f*)A + li);
            const uint64_t ga = (uint64_t)(uintptr_t)(ip + (size_t)gy * IMG + gx);
            asm volatile("global_load_async_to_lds_b128 %0, %1, off"
                         :: "v"(lo), "v"(ga) : "memory");
        } else {
#pragma unroll
            for (int k = 0; k < 4; ++k) {
                const int gxx = gx + k;
                float v = PINF;
                if ((unsigned)gy < (unsigned)IMG && (unsigned)gxx < (unsigned)IMG)
                    v = ip[(size_t)gy * IMG + gxx];
                A[li + k] = v;
            }
        }
    }
#if __has_builtin(__builtin_amdgcn_s_wait_asynccnt)
    __builtin_amdgcn_s_wait_asynccnt(0);
#else
    asm volatile("s_wait_asynccnt 0" ::: "memory");
#endif
    __syncthreads();

    // ---- Stage 1: vertical min, rows r in [1,19), all 34 quads --------------
    for (int u = tid; u < 18 * 34; u += 256) {
        const int r = 1 + u / 34;
        const int g = u - (r - 1) * 34;
        const int base = r * RW + g * 4;
        const v4f a = *(const v4f*)&A[base - RW];
        const v4f b = *(const v4f*)&A[base];
        const v4f c = *(const v4f*)&A[base + RW];
        *(v4f*)&B[base] = vmin3(a, b, c);
    }
    __syncthreads();

    // ---- Stage 2: horizontal min -> erosion --------------------------------
    // Valid region consumed downstream: c in [3,133). Quads cover [0,136);
    // spill reads (B[base-1]/B[base+4]) stay inside the array (adjacent-row
    // garbage) and the garbage columns are never consumed.
    // Image-border tiles additionally force out-of-image positions to -inf
    // (the dilation identity) -- block-uniform branch keeps interior fast.
    const bool edge = (x0 == 0) | (x0 == IMG - TX) | (y0 == 0) | (y0 == IMG - TY);
    if (!edge) {
        for (int u = tid; u < 18 * 34; u += 256) {
            const int r = 1 + u / 34;
            const int g = u - (r - 1) * 34;
            const int base = r * RW + g * 4;
            const v4f q = *(const v4f*)&B[base];
            const float l = B[base - 1], rr = B[base + 4];
            *(v4f*)&A[base] = hmin3(l, q, rr);
        }
    } else {
        for (int u = tid; u < 18 * 34; u += 256) {
            const int r = 1 + u / 34;
            const int g = u - (r - 1) * 34;
            const int base = r * RW + g * 4;
            const v4f q = *(const v4f*)&B[base];
            const float l = B[base - 1], rr = B[base + 4];
            v4f o = hmin3(l, q, rr);
            const int  gy    = y0 - 2 + r;
            const int  gx0   = x0 - 4 + g * 4;
            const bool rowok = (unsigned)gy < (unsigned)IMG;
            o.x = (rowok && (unsigned)(gx0 + 0) < (unsigned)IMG) ? o.x : -PINF;
            o.y = (rowok && (unsigned)(gx0 + 1) < (unsigned)IMG) ? o.y : -PINF;
            o.z = (rowok && (unsigned)(gx0 + 2) < (unsigned)IMG) ? o.z : -PINF;
            o.w = (rowok && (unsigned)(gx0 + 3) < (unsigned)IMG) ? o.w : -PINF;
            *(v4f*)&A[base] = o;
        }
    }
    __syncthreads();

    // ---- Stage 3: vertical max, rows r in [2,18), all 34 quads --------------
    for (int u = tid; u < 16 * 34; u += 256) {
        const int r = 2 + u / 34;
        const int g = u - (r - 2) * 34;
        const int base = r * RW + g * 4;
        const v4f a = *(const v4f*)&A[base - RW];
        const v4f b = *(const v4f*)&A[base];
        const v4f c = *(const v4f*)&A[base + RW];
        *(v4f*)&B[base] = vmax3(a, b, c);
    }
    __syncthreads();

    // ---- Stage 4: horizontal max + nontemporal b128 stores ------------------
    for (int u = tid; u < TY * (TX / 4); u += 256) {
        const int row = u >> 5;             // 0..15
        const int g   = u & 31;             // 0..31
        const int r   = row + 2;
        const int c   = 4 + g * 4;          // x0 .. x0+127, 16B aligned
        const int base = r * RW + c;
        const v4f q = *(const v4f*)&B[base];
        const float l = B[base - 1], rr = B[base + 4];
        const v4f o = hmax3(l, q, rr);
        __builtin_nontemporal_store(o, (v4f*)(op + (size_t)(y0 + row) * IMG + x0 + g * 4));
    }
}

extern "C" void kernel_launch(void* const* d_in, const int* in_sizes, int n_in,
                              void* d_out, int out_size, void* d_ws, size_t ws_size,
                              hipStream_t stream) {
    (void)n_in; (void)out_size; (void)d_ws; (void)ws_size;
    const float* x = (const float*)d_in[0];
    float*       y = (float*)d_out;
    const int planes = in_sizes[0] / (IMG * IMG);   // 16*3 = 48
    dim3 grid(IMG / TX, IMG / TY, planes);          // (8, 64, 48)
    dim3 block(256);
    hipLaunchKernelGGL(MorphOpen_f32_kernel, grid, block, 0, stream, x, y);
}